// GNN_21131239097124
// MI455X (gfx1250) — compile-verified
//
#include <hip/hip_runtime.h>
#include <stdint.h>

#define EMB 64
#define WAVE 32

// CDNA5 async global->LDS path (ASYNCcnt). Guarded: absent builtin -> shfl fallback.
#if defined(__gfx1250__) && __has_builtin(__builtin_amdgcn_global_load_async_to_lds_b64)
#define USE_ASYNC_LDS 1
#else
#define USE_ASYNC_LDS 0
#endif

#if defined(__gfx1250__) && __has_builtin(__builtin_amdgcn_s_wait_asynccnt)
#define WAIT_ASYNC(n) __builtin_amdgcn_s_wait_asynccnt(n)
#elif defined(__gfx1250__)
#define WAIT_ASYNC(n) asm volatile("s_wait_asynccnt %0" ::"i"(n) : "memory")
#else
#define WAIT_ASYNC(n)
#endif

// Exact builtin parameter types (from clang diagnostic):
//   param0: int __attribute__((vector_size(8))) * in AS1 (printed as __device__)
//   param1: LDS-side pointer (AS3)
typedef int v2i32 __attribute__((vector_size(8)));
typedef __attribute__((address_space(1))) v2i32 gas_v2i32;
typedef __attribute__((address_space(3))) v2i32 las_v2i32;

// ---------------- CSR build ----------------

__global__ void deg_kernel(const int* __restrict__ col, int* __restrict__ deg, int E) {
    int e = blockIdx.x * blockDim.x + threadIdx.x;
    if (e < E) atomicAdd(&deg[col[e]], 1);
}

__global__ void dis_kernel(const int* __restrict__ deg, float* __restrict__ dis, int N) {
    int v = blockIdx.x * blockDim.x + threadIdx.x;
    if (v < N) {
        int d = deg[v];
        dis[v] = (d > 0) ? rsqrtf((float)d) : 0.0f;
    }
}

// Per-block inclusive scan of deg into tmp; block sums out.
__global__ void scanA_kernel(const int* __restrict__ deg, int* __restrict__ tmp,
                             int* __restrict__ bsums, int N) {
    __shared__ int lds[1024];
    int t = threadIdx.x;
    int i = blockIdx.x * 1024 + t;
    int v = (i < N) ? deg[i] : 0;
    lds[t] = v;
    __syncthreads();
    for (int off = 1; off < 1024; off <<= 1) {
        int tv = (t >= off) ? lds[t - off] : 0;
        __syncthreads();
        lds[t] += tv;
        __syncthreads();
    }
    if (i < N) tmp[i] = lds[t];
    if (t == 1023) bsums[blockIdx.x] = lds[1023];
}

// Single-block exclusive scan of block sums (nb <= 1024).
__global__ void scanB_kernel(int* __restrict__ bsums, int nb) {
    __shared__ int lds[1024];
    int t = threadIdx.x;
    int v = (t < nb) ? bsums[t] : 0;
    lds[t] = v;
    __syncthreads();
    for (int off = 1; off < 1024; off <<= 1) {
        int tv = (t >= off) ? lds[t - off] : 0;
        __syncthreads();
        lds[t] += tv;
        __syncthreads();
    }
    if (t < nb) bsums[t] = lds[t] - v;  // exclusive
}

__global__ void scanC_kernel(const int* __restrict__ tmp, const int* __restrict__ bsums,
                             int* __restrict__ offs, int N) {
    int i = blockIdx.x * blockDim.x + threadIdx.x;
    if (i < N) offs[i + 1] = tmp[i] + bsums[i >> 10];
    if (i == 0) offs[0] = 0;
}

// Interleaved (row, norm) pairs: one b64 moves a whole edge record.
__global__ void fill_kernel(const int* __restrict__ row, const int* __restrict__ col,
                            const float* __restrict__ dis, const int* __restrict__ offs,
                            int* __restrict__ cursor, int2* __restrict__ csr_pair, int E) {
    int e = blockIdx.x * blockDim.x + threadIdx.x;
    if (e < E) {
        int r = row[e];
        int c = col[e];
        int pos = offs[c] + atomicAdd(&cursor[c], 1);
        csr_pair[pos] = make_int2(r, __float_as_int(dis[r] * dis[c]));
    }
}

// ---------------- SpMM layer: one wave32 per destination node ----------------
// Edge chunks (32 x int2 = 256B) are streamed global->LDS with the CDNA5 async
// path (ASYNCcnt), double-buffered per wave; compute reads them back as
// conflict-free LDS broadcasts (ds_load_b64, uniform address). Each lane owns
// features [2*lane, 2*lane+1]; gathers are coalesced 256B/row. No atomics.

__global__ void __launch_bounds__(256)
spmm_kernel(const int* __restrict__ offs, const int2* __restrict__ csr_pair,
            const float* __restrict__ x_in, float* __restrict__ x_out,
            float* __restrict__ acc, int N) {
    int wid  = (blockIdx.x * blockDim.x + threadIdx.x) >> 5;
    int lane = threadIdx.x & (WAVE - 1);
#if USE_ASYNC_LDS
    __shared__ int2 stage[8][2][WAVE];   // 8 waves x 2 buffers x 32 edges = 4 KB
    int wib = threadIdx.x >> 5;
#endif
    if (wid >= N) return;

    int beg = offs[wid];
    int end = offs[wid + 1];

    float sx = 0.0f, sy = 0.0f;

#if USE_ASYNC_LDS
    int nch = (end - beg + WAVE - 1) / WAVE;
    if (nch > 0) {
        // Prime chunk 0 into buffer 0 (per-lane 8B -> 256B per wave).
        {
            const int2* g = csr_pair + beg + lane;
            __builtin_amdgcn_global_load_async_to_lds_b64(
                (gas_v2i32*)(uintptr_t)g,
                (las_v2i32*)(uintptr_t)&stage[wib][0][lane], 0, 0);
        }
        for (int c = 0; c < nch; ++c) {
            if (c + 1 < nch) {
                const int2* g = csr_pair + beg + (c + 1) * WAVE + lane;
                __builtin_amdgcn_global_load_async_to_lds_b64(
                    (gas_v2i32*)(uintptr_t)g,
                    (las_v2i32*)(uintptr_t)&stage[wib][(c + 1) & 1][lane], 0, 0);
                WAIT_ASYNC(1);   // async ops complete in order: chunk c is ready
            } else {
                WAIT_ASYNC(0);
            }
            int base = beg + c * WAVE;
            int cnt  = end - base;
            if (cnt > WAVE) cnt = WAVE;
            const int2* buf = &stage[wib][c & 1][0];

            // Prefetch this lane's own gather row (both 128B lines) so later
            // k-iterations hit L0 (global_prefetch_b8).
            {
                int2 mine = buf[lane];
                const float* rp = x_in + (size_t)mine.x * EMB;
                __builtin_prefetch(rp, 0, 3);
                __builtin_prefetch(rp + 32, 0, 3);
            }
#pragma unroll 4
            for (int k = 0; k < cnt; ++k) {
                int2  p  = buf[k];                    // LDS broadcast read
                float nn = __int_as_float(p.y);
                const float2 xv =
                    *reinterpret_cast<const float2*>(x_in + (size_t)p.x * EMB + lane * 2);
                sx = fmaf(nn, xv.x, sx);
                sy = fmaf(nn, xv.y, sy);
            }
        }
    }
#else
    for (int j = beg; j < end; j += WAVE) {
        int idx = j + lane;
        int2 p = make_int2(0, 0);
        if (idx < end) p = csr_pair[idx];
        const float* rp = x_in + (size_t)p.x * EMB;
        __builtin_prefetch(rp, 0, 3);
        __builtin_prefetch(rp + 32, 0, 3);
        int cnt = end - j;
        if (cnt > WAVE) cnt = WAVE;
#pragma unroll 4
        for (int k = 0; k < cnt; ++k) {
            int   rr = __shfl(p.x, k, WAVE);
            float nn = __int_as_float(__shfl(p.y, k, WAVE));
            const float2 xv =
                *reinterpret_cast<const float2*>(x_in + (size_t)rr * EMB + lane * 2);
            sx = fmaf(nn, xv.x, sx);
            sy = fmaf(nn, xv.y, sy);
        }
    }
#endif

    size_t o = (size_t)wid * EMB + lane * 2;
    x_out[o]     = sx;
    x_out[o + 1] = sy;
    acc[o]      += sx;   // wave owns node: race-free RMW
    acc[o + 1]  += sy;
}

__global__ void scale_kernel(float* __restrict__ out, float s, int n) {
    int i = blockIdx.x * blockDim.x + threadIdx.x;
    if (i < n) out[i] *= s;
}

// ---------------- driver ----------------

extern "C" void kernel_launch(void* const* d_in, const int* in_sizes, int n_in,
                              void* d_out, int out_size, void* d_ws, size_t ws_size,
                              hipStream_t stream) {
    const float* emb  = (const float*)d_in[0];
    const int*   eidx = (const int*)d_in[1];

    const int N = in_sizes[0] / EMB;   // 100000
    const int E = in_sizes[1] / 2;     // 3200000
    const int* row = eidx;             // edge_index[0]
    const int* col = eidx + E;         // edge_index[1]

    auto al = [](size_t x) { return (x + 255) & ~(size_t)255; };
    char* p = (char*)d_ws;
    int*   deg      = (int*)p;   p += al((size_t)N * 4);
    int*   cursor   = (int*)p;   p += al((size_t)N * 4);
    float* dis      = (float*)p; p += al((size_t)N * 4);
    int*   offs     = (int*)p;   p += al((size_t)(N + 1) * 4);
    int*   tmp      = (int*)p;   p += al((size_t)N * 4);
    int*   bsums    = (int*)p;   p += al((size_t)1024 * 4);
    int2*  csr_pair = (int2*)p;  p += al((size_t)E * 8);
    float* xA       = (float*)p; p += al((size_t)N * EMB * 4);
    float* xB       = (float*)p;

    float* out = (float*)d_out;

    const int TB = 256;
    const int nbScan = (N + 1023) / 1024;

    // Fresh state every call (deterministic; ws is not re-poisoned between replays).
    (void)hipMemsetAsync(deg, 0, (size_t)N * 4, stream);
    (void)hipMemsetAsync(cursor, 0, (size_t)N * 4, stream);
    // acc starts as layer-0 embeddings.
    (void)hipMemcpyAsync(out, emb, (size_t)N * EMB * 4, hipMemcpyDeviceToDevice, stream);

    // CSR-by-destination build.
    deg_kernel<<<(E + TB - 1) / TB, TB, 0, stream>>>(col, deg, E);
    dis_kernel<<<(N + TB - 1) / TB, TB, 0, stream>>>(deg, dis, N);
    scanA_kernel<<<nbScan, 1024, 0, stream>>>(deg, tmp, bsums, N);
    scanB_kernel<<<1, 1024, 0, stream>>>(bsums, nbScan);
    scanC_kernel<<<(N + TB - 1) / TB, TB, 0, stream>>>(tmp, bsums, offs, N);
    fill_kernel<<<(E + TB - 1) / TB, TB, 0, stream>>>(row, col, dis, offs, cursor,
                                                      csr_pair, E);

    // 3 propagation layers, atomic-free, ping-pong x buffers, acc fused.
    const int wavesPerBlock = TB / WAVE;                    // 8 waves
    const int spmmBlocks = (N + wavesPerBlock - 1) / wavesPerBlock;
    spmm_kernel<<<spmmBlocks, TB, 0, stream>>>(offs, csr_pair, emb, xA, out, N);
    spmm_kernel<<<spmmBlocks, TB, 0, stream>>>(offs, csr_pair, xA, xB, out, N);
    spmm_kernel<<<spmmBlocks, TB, 0, stream>>>(offs, csr_pair, xB, xA, out, N);

    // Mean over (NUM_LAYERS + 1) = 4 stacked layers.
    scale_kernel<<<(N * EMB + TB - 1) / TB, TB, 0, stream>>>(out, 0.25f, N * EMB);
}